// RNN_42812234007161
// MI455X (gfx1250) — compile-verified
//
#include <hip/hip_runtime.h>
#include <hip/hip_bf16.h>
#include <math.h>

// RNN: L=2 layers, H=4 hidden, I=O=1, B=8192, S=2048.
// 16 batch rows per wave along the WMMA N dimension; fp32
// V_WMMA_F32_16X16X4_F32 with A = [Wh; 0; Wh; 0] (rows 0-3 and 8-11 = Wh)
// so BOTH half-waves produce bit-identical valid results for the same 16
// batch columns: no cross-lane traffic and no divergence anywhere in the
// serial loop. C operand folds to inline 0; bias+input terms are added
// post-WMMA so the two layer WMMAs issue back-to-back. The x-stream load
// is software-pipelined one 4-step group ahead.

typedef __attribute__((ext_vector_type(2))) float v2f;
typedef __attribute__((ext_vector_type(8))) float v8f;

#define NL 2
#define NH 4
#define NB 8192
#define NS 2048

__device__ __forceinline__ float fast_tanh(float x) {
#if __has_builtin(__builtin_amdgcn_tanhf)
    return __builtin_amdgcn_tanhf(x);   // v_tanh_f32 on gfx1250
#else
    return tanhf(x);
#endif
}

__global__ __launch_bounds__(256) void rnn_wmma_kernel(
    const float* __restrict__ x,    // [B, S, 1]
    const float* __restrict__ Wh,   // [L, H, H]
    const float* __restrict__ bh,   // [L, H]
    const float* __restrict__ Wi,   // [L, H, 1]
    const float* __restrict__ bi,   // [L, H]
    const float* __restrict__ Wo,   // [L, 1, H]
    const float* __restrict__ bo,   // [L, 1]
    float* __restrict__ out,        // [S*B] flat: out[t*B + b] (raw reshape)
    float* __restrict__ hfin)       // [L, B, H]
{
    const int tid  = blockIdx.x * blockDim.x + threadIdx.x;
    const int lane = threadIdx.x & 31;
    const int wave = tid >> 5;
    const int tile = wave << 4;       // 16 batch rows per wave
    const int n    = lane & 15;       // batch column within tile (= A row M)
    const bool lo  = lane < 16;

    // ---- A matrices in WMMA A layout (16x4 f32, 2 VGPRs) ----
    // lane L: M = L%16; VGPR0: K = (L<16?0:2), VGPR1: K = (L<16?1:3)
    // Rows 0-3 AND 8-11 hold Wh[l] (duplicate) so lanes 16-31 (D rows 8-11)
    // compute the same valid update as lanes 0-15. Rows 4-7, 12-15 are zero.
    const int kb = lo ? 0 : 2;
    v2f amat0 = {0.0f, 0.0f};
    v2f amat1 = {0.0f, 0.0f};
    if ((n & 4) == 0) {               // M in {0..3, 8..11}
        const int r = n & 3;          // Wh row
        amat0.x = Wh[0 * NH * NH + r * NH + kb];
        amat0.y = Wh[0 * NH * NH + r * NH + kb + 1];
        amat1.x = Wh[1 * NH * NH + r * NH + kb];
        amat1.y = Wh[1 * NH * NH + r * NH + kb + 1];
    }

    // ---- wave-uniform scalars (s_loads) ----
    float cb0[NH], cb1[NH], wi0[NH], wi1[NH], wo0[NH], wo1[NH];
#pragma unroll
    for (int v = 0; v < NH; ++v) {
        cb0[v] = bh[v] + bi[v];
        cb1[v] = bh[NH + v] + bi[NH + v];
        wi0[v] = Wi[v];
        wi1[v] = Wi[NH + v];
        wo0[v] = Wo[v];
        wo1[v] = Wo[NH + v];
    }
    const float bo0 = bo[0];
    const float bo1 = bo[1];

    // ---- per-lane hidden state (valid in ALL lanes) ----
    float h00 = 0.f, h01 = 0.f, h02 = 0.f, h03 = 0.f;   // layer 0
    float h10 = 0.f, h11 = 0.f, h12 = 0.f, h13 = 0.f;   // layer 1

    const v8f zc = {0.f, 0.f, 0.f, 0.f, 0.f, 0.f, 0.f, 0.f};  // folds to inline 0

    const float* xin   = x + (size_t)(tile + n) * NS;   // this lane's batch row
    const float4* xin4 = reinterpret_cast<const float4*>(xin);
    float* outp = out + (tile + n);

    // software pipeline: group t0 is loaded one iteration ahead
    float4 cur = xin4[0];

    for (int t0 = 0; t0 < NS; t0 += 4) {
        // issue next group's load now; clamp keeps the address in-bounds
        const int nidx = (t0 + 4 < NS ? t0 + 4 : NS - 4) >> 2;
        const float4 nxt = xin4[nidx];
        __builtin_prefetch(xin + t0 + 128, 0, 1);  // global_prefetch_b8, 512B ahead
#pragma unroll
        for (int j = 0; j < 4; ++j) {
            const float xt = (j == 0) ? cur.x : (j == 1) ? cur.y
                             : (j == 2) ? cur.z : cur.w;

            // B operands (4x16, batch along N): v0={K0|K2}, v1={K1|K3}.
            // Upper half uses its own local copies of h2/h3 -> no shuffles.
            v2f b0, b1;
            b0.x = lo ? h00 : h02;  b0.y = lo ? h01 : h03;
            b1.x = lo ? h10 : h12;  b1.y = lo ? h11 : h13;

            // Both recurrence matmuls issue up front (no same-step deps).
            v8f d0 = __builtin_amdgcn_wmma_f32_16x16x4_f32(
                false, amat0, false, b0, (short)0, zc, false, false);
            v8f d1 = __builtin_amdgcn_wmma_f32_16x16x4_f32(
                false, amat1, false, b1, (short)0, zc, false, false);

            // ---- layer 0 post: inner = Wh*h + (bh+bi + x*Wi); tanh ----
            h00 = fast_tanh(d0[0] + __builtin_fmaf(xt, wi0[0], cb0[0]));
            h01 = fast_tanh(d0[1] + __builtin_fmaf(xt, wi0[1], cb0[1]));
            h02 = fast_tanh(d0[2] + __builtin_fmaf(xt, wi0[2], cb0[2]));
            h03 = fast_tanh(d0[3] + __builtin_fmaf(xt, wi0[3], cb0[3]));

            float tok = bo0;
            tok = __builtin_fmaf(h00, wo0[0], tok);
            tok = __builtin_fmaf(h01, wo0[1], tok);
            tok = __builtin_fmaf(h02, wo0[2], tok);
            tok = __builtin_fmaf(h03, wo0[3], tok);

            // ---- layer 1 post ----
            h10 = fast_tanh(d1[0] + __builtin_fmaf(tok, wi1[0], cb1[0]));
            h11 = fast_tanh(d1[1] + __builtin_fmaf(tok, wi1[1], cb1[1]));
            h12 = fast_tanh(d1[2] + __builtin_fmaf(tok, wi1[2], cb1[2]));
            h13 = fast_tanh(d1[3] + __builtin_fmaf(tok, wi1[3], cb1[3]));

            float o = bo1;
            o = __builtin_fmaf(h10, wo1[0], o);
            o = __builtin_fmaf(h11, wo1[1], o);
            o = __builtin_fmaf(h12, wo1[2], o);
            o = __builtin_fmaf(h13, wo1[3], o);

            // Unconditional store: lane n and lane n+16 write the SAME value
            // to the same address (both halves are bit-identical), so the
            // loop body has zero exec divergence.
            outp[(size_t)(t0 + j) * NB] = o;   // out[t*B + b]
        }
        cur = nxt;
    }

    // ---- h_final: [L, B, H]; both halves valid -> fully converged store:
    // lanes 0-15 store layer 0, lanes 16-31 store layer 1.
    {
        const float4 f = lo ? make_float4(h00, h01, h02, h03)
                            : make_float4(h10, h11, h12, h13);
        float4* hf = reinterpret_cast<float4*>(hfin);
        hf[(lo ? 0 : NB) + tile + n] = f;
    }
}

extern "C" void kernel_launch(void* const* d_in, const int* in_sizes, int n_in,
                              void* d_out, int out_size, void* d_ws, size_t ws_size,
                              hipStream_t stream) {
    (void)in_sizes; (void)n_in; (void)out_size; (void)d_ws; (void)ws_size;
    const float* x  = (const float*)d_in[0];
    const float* Wh = (const float*)d_in[1];
    const float* bh = (const float*)d_in[2];
    const float* Wi = (const float*)d_in[3];
    const float* bi = (const float*)d_in[4];
    const float* Wo = (const float*)d_in[5];
    const float* bo = (const float*)d_in[6];
    float* out  = (float*)d_out;
    float* hfin = out + (size_t)NB * NS;   // outputs then h_final, flat

    // 8192/16 = 512 waves = 16384 threads; 64 blocks x 256 threads (8 waves)
    const int threads = (NB / 16) * 32;
    rnn_wmma_kernel<<<threads / 256, 256, 0, stream>>>(x, Wh, bh, Wi, bi, Wo, bo,
                                                       out, hfin);
}